// RVLinearlayer_46205258170414
// MI455X (gfx1250) — compile-verified
//
#include <hip/hip_runtime.h>
#include <hip/hip_bf16.h>

typedef __attribute__((ext_vector_type(2))) float v2f;
typedef __attribute__((ext_vector_type(8))) float v8f;
typedef __attribute__((ext_vector_type(4))) unsigned int u32x4;
typedef __attribute__((ext_vector_type(8))) int i32x8;
typedef __attribute__((ext_vector_type(4))) int i32x4;

#define B_   64
#define DIN  512
#define DOUT 256
#define KB   128                 // k-block staged per TDM tile
#define AROW 129                 // LDS row stride in dwords (128 data + 1 TDM pad)

// d_out layout: [0,16384) mu_out (b*256+o), [16384, +64*256*256) Sigma_out
// d_ws  layout: diagacc 64*256 f32 | T 64*512*256 f32 | S 64*512*512 f32

__device__ __forceinline__ float softplus_f(float x) {
    return (x > 20.0f) ? x : log1pf(__expf(x));
}

__global__ void zero_diag_kernel(float* __restrict__ diagacc) {
    int idx = blockIdx.x * blockDim.x + threadIdx.x;
    if (idx < B_ * DOUT) diagacc[idx] = 0.0f;
}

// ---------------------------------------------------------------------------
// S[b][i][k] = sigma[b][k][i] + mu[b][i]*mu[b][k]   (LDS 32x33 tile transpose)
// ---------------------------------------------------------------------------
__global__ __launch_bounds__(256) void build_s_kernel(const float* __restrict__ sigma,
                                                      const float* __restrict__ mu,
                                                      float* __restrict__ S) {
    __shared__ float tile[32][33];
    const int b  = blockIdx.z;
    const int i0 = blockIdx.y * 32;
    const int k0 = blockIdx.x * 32;
    const int tx = threadIdx.x & 31;
    const int ty = threadIdx.x >> 5;          // 0..7

    const float* sb = sigma + (size_t)b * DIN * DIN;
#pragma unroll
    for (int r = 0; r < 4; ++r) {
        const int kr = ty + r * 8;            // k offset within tile
        tile[kr][tx] = sb[(size_t)(k0 + kr) * DIN + i0 + tx];
    }
    __syncthreads();
    const float* mub = mu + (size_t)b * DIN;
    float* Sb = S + (size_t)b * DIN * DIN;
#pragma unroll
    for (int r = 0; r < 4; ++r) {
        const int ir = ty + r * 8;            // i offset within tile
        Sb[(size_t)(i0 + ir) * DIN + k0 + tx] =
            tile[tx][ir] + mub[i0 + ir] * mub[k0 + tx];
    }
}

// ---------------------------------------------------------------------------
// TDM: load a 16-row x 128-col f32 tile (row stride = rstride elements) into
// LDS at byte offset lds_byte, padding 1 dword after each 128-dword row.
// D# packing per CDNA5 ISA ch.8.3/8.4.
// ---------------------------------------------------------------------------
#if __has_builtin(__builtin_amdgcn_tensor_load_to_lds)
#define HAVE_TDM 1
__device__ __forceinline__ void tdm_load_16x128(unsigned lds_byte,
                                                const float* __restrict__ gptr,
                                                unsigned rstride_elems) {
    const unsigned long long ga = (unsigned long long)(uintptr_t)gptr;
    u32x4 g0;
    g0[0] = 1u;                                           // count=1, user descriptor
    g0[1] = lds_byte;                                     // lds_addr
    g0[2] = (unsigned)(ga & 0xFFFFFFFFu);                 // global_addr[31:0]
    g0[3] = (unsigned)((ga >> 32) & 0x01FFFFFFu) | (2u << 30); // addr[56:32] | type=2
    i32x8 g1;
    g1[0] = (int)((2u << 16) | (1u << 20) | (6u << 22));  // data_size=4B, pad_en, interval=128dw, amount=1dw
    g1[1] = (int)(((unsigned)KB & 0xFFFFu) << 16);        // tensor_dim0 = 128
    g1[2] = (int)(16u << 16);                             // tensor_dim1 = 16
    g1[3] = (int)(((unsigned)KB & 0xFFFFu) << 16);        // tile_dim0 = 128
    g1[4] = (int)16;                                      // tile_dim1 = 16, tile_dim2 = 0
    g1[5] = (int)rstride_elems;                           // tensor_dim0_stride lo32
    g1[6] = 0;                                            // stride hi | dim1_stride lo
    g1[7] = 0;
    const i32x4 z4 = {0, 0, 0, 0};
#if defined(__clang_major__) && (__clang_major__ >= 23)
    const i32x8 z8 = {0, 0, 0, 0, 0, 0, 0, 0};
    __builtin_amdgcn_tensor_load_to_lds(g0, g1, z4, z4, z8, 0);
#else
    __builtin_amdgcn_tensor_load_to_lds(g0, g1, z4, z4, 0);
#endif
}
#else
#define HAVE_TDM 0
#endif

// ---------------------------------------------------------------------------
// T[b] (512x256) = Sigma_in[b] (512x512) @ W (512x256), fp32 WMMA 16x16x4
// ---------------------------------------------------------------------------
__global__ __launch_bounds__(32) void gemm1_kernel(const float* __restrict__ sigma,
                                                   const float* __restrict__ w,
                                                   float* __restrict__ T) {
    const int nt = blockIdx.x, mt = blockIdx.y, b = blockIdx.z;
    const int lane = threadIdx.x;
    const int half = lane >> 4, l15 = lane & 15;

    const float* Sb = sigma + (size_t)b * DIN * DIN;
    const int mrow = mt * 16 + l15;
    const int ncol = nt * 16 + l15;

    v8f c = {};
    for (int k0 = 0; k0 < DIN; k0 += 4) {
        const int ka = k0 + 2 * half;
        v2f a, bb;
        a.x = Sb[(size_t)mrow * DIN + ka];
        a.y = Sb[(size_t)mrow * DIN + ka + 1];
        bb.x = w[(size_t)ka * DOUT + ncol];
        bb.y = w[(size_t)(ka + 1) * DOUT + ncol];
        c = __builtin_amdgcn_wmma_f32_16x16x4_f32(false, a, false, bb, (short)0, c, false, false);
    }
    float* Tb = T + (size_t)b * DIN * DOUT;
#pragma unroll
    for (int r = 0; r < 8; ++r)
        Tb[(size_t)(mt * 16 + r + 8 * half) * DOUT + ncol] = c[r];
}

// ---------------------------------------------------------------------------
// M[b] (256x256) = W^T (256x512) @ T[b] (512x256) -> Sigma_out
// ---------------------------------------------------------------------------
__global__ __launch_bounds__(32) void gemm2_kernel(const float* __restrict__ w,
                                                   const float* __restrict__ T,
                                                   float* __restrict__ out) {
    const int nt = blockIdx.x, mt = blockIdx.y, b = blockIdx.z;
    const int lane = threadIdx.x;
    const int half = lane >> 4, l15 = lane & 15;

    const float* Tb = T + (size_t)b * DIN * DOUT;
    const int mrow = mt * 16 + l15;
    const int ncol = nt * 16 + l15;

    v8f c = {};
    for (int k0 = 0; k0 < DIN; k0 += 4) {
        const int ka = k0 + 2 * half;
        v2f a, bb;
        a.x = w[(size_t)ka * DOUT + mrow];            // A[m][k] = W[k][m]
        a.y = w[(size_t)(ka + 1) * DOUT + mrow];
        bb.x = Tb[(size_t)ka * DOUT + ncol];
        bb.y = Tb[(size_t)(ka + 1) * DOUT + ncol];
        c = __builtin_amdgcn_wmma_f32_16x16x4_f32(false, a, false, bb, (short)0, c, false, false);
    }
    float* Sout = out + 16384;
#pragma unroll
    for (int r = 0; r < 8; ++r)
        Sout[(size_t)b * DOUT * DOUT + (size_t)(mt * 16 + r + 8 * half) * DOUT + ncol] = c[r];
}

// ---------------------------------------------------------------------------
// diag[b,o] = sum_{i,k} softplus(w_sigma[o,i,k]) * S[b,i,k]
// A-panel (w_sigma 16 o-rows x 128 k) streamed via TDM into LDS, double
// buffered on TENSORcnt; B read contiguously from pre-transposed S.
// Grid: x = o-tile (16), y = i-split (64 x 8 rows). One wave per block.
// ---------------------------------------------------------------------------
__global__ __launch_bounds__(32) void diag_kernel(const float* __restrict__ wsig,
                                                  const float* __restrict__ S,
                                                  float* __restrict__ diagacc) {
    __shared__ float atile[2][16 * AROW];
    const int ot = blockIdx.x;   // 0..15
    const int sp = blockIdx.y;   // 0..63
    const int lane = threadIdx.x;
    const int half = lane >> 4, l15 = lane & 15;
    const int i0 = sp * 8;

    const float* abase = wsig + (size_t)(ot * 16) * (DIN * DIN);
    const unsigned lds0 = (unsigned)(uintptr_t)&atile[0][0];   // low 32 bits = LDS offset
    const unsigned lds1 = (unsigned)(uintptr_t)&atile[1][0];

#if HAVE_TDM
    // tile tidx: i = i0 + tidx/4, kb = (tidx%4)*KB
    tdm_load_16x128(lds0, abase + (size_t)i0 * DIN, DIN * DIN);
#else
    for (int idx = lane; idx < 16 * KB; idx += 32) {
        const int r = idx >> 7, cc = idx & (KB - 1);
        atile[0][r * AROW + cc] = abase[(size_t)r * DIN * DIN + (size_t)i0 * DIN + cc];
    }
#endif

    v8f c[4] = {{}, {}, {}, {}};

    for (int tidx = 0; tidx < 32; ++tidx) {
        const int buf = tidx & 1;
        const int i  = i0 + (tidx >> 2);
        const int kb = (tidx & 3) * KB;
        // prefetch next tile into the other buffer
        if (tidx + 1 < 32) {
            const int ni  = i0 + ((tidx + 1) >> 2);
            const int nkb = ((tidx + 1) & 3) * KB;
#if HAVE_TDM
            tdm_load_16x128(buf ? lds0 : lds1,
                            abase + (size_t)ni * DIN + nkb, DIN * DIN);
#else
            float* dst = &atile[buf ^ 1][0];
            for (int idx = lane; idx < 16 * KB; idx += 32) {
                const int r = idx >> 7, cc = idx & (KB - 1);
                dst[r * AROW + cc] =
                    abase[(size_t)r * DIN * DIN + (size_t)ni * DIN + nkb + cc];
            }
#endif
        }
#if HAVE_TDM
        if (tidx + 1 < 32) __builtin_amdgcn_s_wait_tensorcnt(1);
        else               __builtin_amdgcn_s_wait_tensorcnt(0);
        // Compiler fence: the TDM builtin writes LDS through an integer
        // address the alias analysis cannot see; force the atile loads below
        // to be materialized after the hardware wait.
        asm volatile("" ::: "memory");
#else
        __builtin_amdgcn_s_barrier_signal(-1);
        __builtin_amdgcn_s_barrier_wait(-1);
#endif
        const float* arow = &atile[buf][l15 * AROW];
#pragma unroll 2
        for (int k0 = 0; k0 < KB; k0 += 4) {
            const int ka = k0 + 2 * half;
            v2f a;
            a.x = softplus_f(arow[ka]);
            a.y = softplus_f(arow[ka + 1]);
#pragma unroll
            for (int t = 0; t < 4; ++t) {
                const int bg = t * 16 + l15;
                const float* srow = S + (size_t)bg * DIN * DIN + (size_t)i * DIN + kb;
                v2f bb;
                bb.x = srow[ka];
                bb.y = srow[ka + 1];
                c[t] = __builtin_amdgcn_wmma_f32_16x16x4_f32(false, a, false, bb,
                                                             (short)0, c[t], false, false);
            }
        }
    }

#pragma unroll
    for (int t = 0; t < 4; ++t) {
        const int bg = t * 16 + l15;
#pragma unroll
        for (int r = 0; r < 8; ++r) {
            const int o = ot * 16 + r + 8 * half;
            atomicAdd(&diagacc[bg * DOUT + o], c[t][r]);
        }
    }
}

// ---------------------------------------------------------------------------
// mu_out[b,o] = sum_i w[i,o]*mu[b,i] + b_mu[o]; Sigma_out[b,o,o] += diagacc
// ---------------------------------------------------------------------------
__global__ void finalize_kernel(const float* __restrict__ w,
                                const float* __restrict__ mu,
                                const float* __restrict__ bmu,
                                const float* __restrict__ diagacc,
                                float* __restrict__ out) {
    const int idx = blockIdx.x * blockDim.x + threadIdx.x;
    if (idx >= B_ * DOUT) return;
    const int b = idx >> 8, o = idx & 255;
    const float* mub = mu + (size_t)b * DIN;
    float acc = bmu[o];
    for (int i = 0; i < DIN; ++i) acc += w[(size_t)i * DOUT + o] * mub[i];
    out[idx] = acc;
    out[16384 + (size_t)b * DOUT * DOUT + (size_t)o * (DOUT + 1)] += diagacc[idx];
}

extern "C" void kernel_launch(void* const* d_in, const int* in_sizes, int n_in,
                              void* d_out, int out_size, void* d_ws, size_t ws_size,
                              hipStream_t stream) {
    const float* mu_in    = (const float*)d_in[0];   // [64,512,1]
    const float* sigma_in = (const float*)d_in[1];   // [64,512,512]
    const float* w_mu     = (const float*)d_in[2];   // [512,256]
    const float* w_sigma  = (const float*)d_in[3];   // [256,512,512]
    const float* b_mu     = (const float*)d_in[4];   // [256,1]
    float* out = (float*)d_out;

    float* diagacc = (float*)d_ws;                        // 64*256
    float* T       = diagacc + B_ * DOUT;                 // 64*512*256
    float* S       = T + (size_t)B_ * DIN * DOUT;         // 64*512*512

    zero_diag_kernel<<<dim3(64), dim3(256), 0, stream>>>(diagacc);
    build_s_kernel<<<dim3(16, 16, 64), dim3(256), 0, stream>>>(sigma_in, mu_in, S);
    gemm1_kernel<<<dim3(16, 32, 64), dim3(32), 0, stream>>>(sigma_in, w_mu, T);
    gemm2_kernel<<<dim3(16, 16, 64), dim3(32), 0, stream>>>(w_mu, T, out);
    diag_kernel<<<dim3(16, 64), dim3(32), 0, stream>>>(w_sigma, S, diagacc);
    finalize_kernel<<<dim3(64), dim3(256), 0, stream>>>(w_mu, mu_in, b_mu, diagacc, out);
}